// Intention_20478404067361
// MI455X (gfx1250) — compile-verified
//
#include <hip/hip_runtime.h>
#include <math.h>

// ---------------------------------------------------------------------------
// Types for CDNA5 WMMA f32 16x16x4
// ---------------------------------------------------------------------------
typedef float v2f __attribute__((ext_vector_type(2)));
typedef float v8f __attribute__((ext_vector_type(8)));

__device__ __forceinline__ v8f wmma_f32(v2f a, v2f b, v8f c) {
    // D = A(16x4) * B(4x16) + C(16x16), fp32 throughout.
    return __builtin_amdgcn_wmma_f32_16x16x4_f32(
        /*neg_a=*/false, a, /*neg_b=*/false, b,
        /*c_mod=*/(short)0, c, /*reuse_a=*/false, /*reuse_b=*/false);
}

#define BDIM   8
#define NHEADS 16
#define SEQ    1024
#define CDIM   1024
#define HD     64
#define BH     (BDIM * NHEADS)   // 128

// ---------------------------------------------------------------------------
// Generic NT GEMM: C[m][n] = sum_k A[m][k] * B[n][k]  (+ bias[n] in mode 0)
// Register-blocked: each wave computes a 32x64 tile (2 A-frags x 4 B-frags,
// 8 accumulators -> 8 WMMAs per 6 fragment loads, ~10.7 flops/byte).
// Block = 8 waves stacked in M (256 rows x 64 cols per block).
// grid = (M/256, N/64).  mode 0: row-major [M][N] + bias.
// mode 1: scatter to headed layout [b][h][n][d] (m = b*1024+n, col = h*64+d)
// ---------------------------------------------------------------------------
__global__ void __launch_bounds__(256)
gemm_nt_kernel(const float* __restrict__ A, const float* __restrict__ B,
               const float* __restrict__ bias, float* __restrict__ C,
               int M, int N, int Kd, int mode)
{
    const int wave = threadIdx.x >> 5;
    const int lane = threadIdx.x & 31;
    const int hi   = lane >> 4;       // 0 for lanes 0-15, 1 for 16-31
    const int lx   = lane & 15;

    const int rbase = (blockIdx.x * 8 + wave) * 32;  // 32 M rows per wave
    const int cbase = blockIdx.y * 64;               // 64 N cols per wave

    const float* ap0 = A + ((size_t)(rbase + lx)) * Kd + 2 * hi;
    const float* ap1 = ap0 + (size_t)16 * Kd;
    const float* bp0 = B + ((size_t)(cbase + lx)) * Kd + 2 * hi;
    const float* bp1 = bp0 + (size_t)16 * Kd;
    const float* bp2 = bp0 + (size_t)32 * Kd;
    const float* bp3 = bp0 + (size_t)48 * Kd;

    v8f acc[2][4];
#pragma unroll
    for (int i = 0; i < 2; ++i)
#pragma unroll
        for (int t = 0; t < 4; ++t) acc[i][t] = (v8f){};

#pragma unroll 2
    for (int k = 0; k < Kd; k += 4) {
        const v2f a0 = *(const v2f*)(ap0 + k);
        const v2f a1 = *(const v2f*)(ap1 + k);
        const v2f b0 = *(const v2f*)(bp0 + k);
        const v2f b1 = *(const v2f*)(bp1 + k);
        const v2f b2 = *(const v2f*)(bp2 + k);
        const v2f b3 = *(const v2f*)(bp3 + k);
        acc[0][0] = wmma_f32(a0, b0, acc[0][0]);
        acc[0][1] = wmma_f32(a0, b1, acc[0][1]);
        acc[0][2] = wmma_f32(a0, b2, acc[0][2]);
        acc[0][3] = wmma_f32(a0, b3, acc[0][3]);
        acc[1][0] = wmma_f32(a1, b0, acc[1][0]);
        acc[1][1] = wmma_f32(a1, b1, acc[1][1]);
        acc[1][2] = wmma_f32(a1, b2, acc[1][2]);
        acc[1][3] = wmma_f32(a1, b3, acc[1][3]);
    }

    if (mode == 0) {
#pragma unroll
        for (int t = 0; t < 4; ++t) {
            const int col = cbase + t * 16 + lx;
            const float bv = bias ? bias[col] : 0.0f;
#pragma unroll
            for (int i = 0; i < 2; ++i) {
#pragma unroll
                for (int r = 0; r < 8; ++r) {
                    const int row = rbase + i * 16 + r + 8 * hi;
                    C[(size_t)row * N + col] = acc[i][t][r] + bv;
                }
            }
        }
    } else {
#pragma unroll
        for (int t = 0; t < 4; ++t) {
            const int col = cbase + t * 16 + lx;
            const int h = col >> 6, d = col & 63;
#pragma unroll
            for (int i = 0; i < 2; ++i) {
#pragma unroll
                for (int r = 0; r < 8; ++r) {
                    const int row = rbase + i * 16 + r + 8 * hi;  // b*1024 + n
                    const int b = row >> 10, n = row & 1023;
                    C[((((size_t)b * NHEADS + h) * SEQ) + n) * HD + d] = acc[i][t][r];
                }
            }
        }
    }
}

// ---------------------------------------------------------------------------
// scores[bh][q][k] = dot(Q[bh][q][:], K[bh][k][:]) / 8
// Q, K: [BH][SEQ][HD].  Same 32x64 register blocking, K=64 fully unrolled.
// grid = (BH, SEQ/256, SEQ/64), block = 256 (8 waves stacked in q).
// ---------------------------------------------------------------------------
__global__ void __launch_bounds__(256)
scores_kernel(const float* __restrict__ Q, const float* __restrict__ Kt,
              float* __restrict__ trad)
{
    const int bh   = blockIdx.x;
    const int wave = threadIdx.x >> 5;
    const int lane = threadIdx.x & 31;
    const int hi   = lane >> 4;
    const int lx   = lane & 15;

    const int qbase = (blockIdx.y * 8 + wave) * 32;
    const int kbase = blockIdx.z * 64;

    const float* qp0 = Q  + ((size_t)bh * SEQ + qbase + lx) * HD + 2 * hi;
    const float* qp1 = qp0 + (size_t)16 * HD;
    const float* kp0 = Kt + ((size_t)bh * SEQ + kbase + lx) * HD + 2 * hi;
    const float* kp1 = kp0 + (size_t)16 * HD;
    const float* kp2 = kp0 + (size_t)32 * HD;
    const float* kp3 = kp0 + (size_t)48 * HD;

    v8f acc[2][4];
#pragma unroll
    for (int i = 0; i < 2; ++i)
#pragma unroll
        for (int t = 0; t < 4; ++t) acc[i][t] = (v8f){};

#pragma unroll
    for (int k = 0; k < HD; k += 4) {
        const v2f a0 = *(const v2f*)(qp0 + k);
        const v2f a1 = *(const v2f*)(qp1 + k);
        const v2f b0 = *(const v2f*)(kp0 + k);
        const v2f b1 = *(const v2f*)(kp1 + k);
        const v2f b2 = *(const v2f*)(kp2 + k);
        const v2f b3 = *(const v2f*)(kp3 + k);
        acc[0][0] = wmma_f32(a0, b0, acc[0][0]);
        acc[0][1] = wmma_f32(a0, b1, acc[0][1]);
        acc[0][2] = wmma_f32(a0, b2, acc[0][2]);
        acc[0][3] = wmma_f32(a0, b3, acc[0][3]);
        acc[1][0] = wmma_f32(a1, b0, acc[1][0]);
        acc[1][1] = wmma_f32(a1, b1, acc[1][1]);
        acc[1][2] = wmma_f32(a1, b2, acc[1][2]);
        acc[1][3] = wmma_f32(a1, b3, acc[1][3]);
    }

    float* out = trad + (size_t)bh * SEQ * SEQ;
#pragma unroll
    for (int t = 0; t < 4; ++t) {
        const int col = kbase + t * 16 + lx;
#pragma unroll
        for (int i = 0; i < 2; ++i) {
#pragma unroll
            for (int r = 0; r < 8; ++r) {
                const int qrow = qbase + i * 16 + r + 8 * hi;
                out[(size_t)qrow * SEQ + col] = acc[i][t][r] * 0.125f; // 1/sqrt(64)
            }
        }
    }
}

// ---------------------------------------------------------------------------
// In-place softmax over the QUERY axis (axis=-2): for each (bh, col),
// softmax over the 1024 rows. Coalesced: adjacent threads = adjacent cols.
// grid = (BH, SEQ/256), block = 256
// ---------------------------------------------------------------------------
__global__ void __launch_bounds__(256)
colsoftmax_kernel(float* __restrict__ t)
{
    const size_t bh  = blockIdx.x;
    const int    col = blockIdx.y * 256 + threadIdx.x;
    float* p = t + bh * (size_t)SEQ * SEQ + col;

    float m = -INFINITY;
    for (int r = 0; r < SEQ; ++r) m = fmaxf(m, p[(size_t)r * SEQ]);
    float s = 0.0f;
    for (int r = 0; r < SEQ; ++r) s += __expf(p[(size_t)r * SEQ] - m);
    const float inv = 1.0f / s;
    for (int r = 0; r < SEQ; ++r) {
        p[(size_t)r * SEQ] = __expf(p[(size_t)r * SEQ] - m) * inv;
    }
}

// ---------------------------------------------------------------------------
// Per-(b,h) solve: kk = K^T K + alpha*I (64x64); inv = kk^-1 (Gauss-Jordan,
// SPD so no pivoting); m = inv @ (K^T V); maps = softmax(m, axis=-2).
// One 256-thread block per bh. LDS phases overlap: aug region doubles as
// the K/V staging tiles during accumulation.
// ---------------------------------------------------------------------------
__global__ void __launch_bounds__(256)
solve_kernel(const float* __restrict__ Kin, const float* __restrict__ Vin,
             const float* __restrict__ alpha, float* __restrict__ maps)
{
    __shared__ float aug[64 * 128];   // phase1: sK = aug[0:4096], sV = aug[4096:8192]
    __shared__ float sM[64 * 64];     // ktv, later m
    __shared__ float fac[64];
    __shared__ float pivinv;

    const int tid   = threadIdx.x;
    const int bh    = blockIdx.x;
    const int e     = tid & 63;        // owned column
    const int dbase = (tid >> 6) * 16; // owned row block

    float kk[16], kv[16];
#pragma unroll
    for (int i = 0; i < 16; ++i) { kk[i] = 0.0f; kv[i] = 0.0f; }

    const float* Kb = Kin + (size_t)bh * SEQ * HD;
    const float* Vb = Vin + (size_t)bh * SEQ * HD;
    float* sK = aug;
    float* sV = aug + 4096;

    for (int c = 0; c < SEQ / 64; ++c) {
        for (int idx = tid; idx < 64 * 64; idx += 256) {
            sK[idx] = Kb[(size_t)c * 64 * HD + idx];
            sV[idx] = Vb[(size_t)c * 64 * HD + idx];
        }
        __syncthreads();
        for (int n = 0; n < 64; ++n) {
            const float ke = sK[n * 64 + e];
            const float ve = sV[n * 64 + e];
#pragma unroll
            for (int i = 0; i < 16; ++i) {
                const float kd = sK[n * 64 + dbase + i];
                kk[i] = fmaf(kd, ke, kk[i]);
                kv[i] = fmaf(kd, ve, kv[i]);
            }
        }
        __syncthreads();
    }

    // Build augmented [kk + a*I | I] and stash ktv
    const float a0 = alpha[0];
#pragma unroll
    for (int i = 0; i < 16; ++i) {
        const int d = dbase + i;
        aug[d * 128 + e]      = kk[i] + (d == e ? a0 : 0.0f);
        aug[d * 128 + 64 + e] = (d == e) ? 1.0f : 0.0f;
        sM[d * 64 + e]        = kv[i];
    }
    __syncthreads();

    // Gauss-Jordan, no pivoting (kk is SPD + alpha*I)
    for (int p = 0; p < 64; ++p) {
        if (tid == 0) pivinv = 1.0f / aug[p * 128 + p];
        __syncthreads();
        if (tid < 128) {
            aug[p * 128 + tid] *= pivinv;
        } else if (tid < 192) {
            const int r = tid - 128;
            fac[r] = (r == p) ? 0.0f : aug[r * 128 + p];
        }
        __syncthreads();
        for (int idx = tid; idx < 64 * 128; idx += 256) {
            const int r = idx >> 7, cc = idx & 127;
            if (r != p) aug[idx] = fmaf(-fac[r], aug[p * 128 + cc], aug[idx]);
        }
        __syncthreads();
    }

    // m = inv @ ktv  (inv lives in aug[:,64:128])
    float mv[16];
#pragma unroll
    for (int i = 0; i < 16; ++i) {
        const int d = dbase + i;
        float s = 0.0f;
        for (int j = 0; j < 64; ++j)
            s = fmaf(aug[d * 128 + 64 + j], sM[j * 64 + e], s);
        mv[i] = s;
    }
    __syncthreads();
#pragma unroll
    for (int i = 0; i < 16; ++i) sM[(dbase + i) * 64 + e] = mv[i];
    __syncthreads();

    // softmax over axis=-2 (over d, per column e); then write maps[bh][d][e]
    if (tid < 64) {
        const int ec = tid;
        float mmax = -INFINITY;
        for (int d = 0; d < 64; ++d) mmax = fmaxf(mmax, sM[d * 64 + ec]);
        float s = 0.0f;
        for (int d = 0; d < 64; ++d) s += __expf(sM[d * 64 + ec] - mmax);
        const float inv = 1.0f / s;
        float* outp = maps + (size_t)bh * HD * HD;
        for (int d = 0; d < 64; ++d)
            outp[d * 64 + ec] = __expf(sM[d * 64 + ec] - mmax) * inv;
    }
}

// ---------------------------------------------------------------------------
// out[b][n][h*64+e] = sum_d Q[bh][n][d] * maps[bh][d][e]
// Map tile staged in LDS; each wave: 16 rows x 64 cols (4 WMMA col tiles,
// A-fragment reused). grid = (BH, SEQ/128), block = 256.
// ---------------------------------------------------------------------------
__global__ void __launch_bounds__(256)
qmap_kernel(const float* __restrict__ Q, const float* __restrict__ maps,
            float* __restrict__ outbnc)
{
    __shared__ float sMap[64 * 64];
    const int bh = blockIdx.x;
    const int tid = threadIdx.x;

    const float* mp = maps + (size_t)bh * HD * HD;
    for (int idx = tid; idx < 64 * 64; idx += 256) sMap[idx] = mp[idx];
    __syncthreads();

    const int wave = tid >> 5, lane = tid & 31;
    const int hi = lane >> 4, lx = lane & 15;
    const int nrow = blockIdx.y * 128 + wave * 16 + lx;
    const float* qp = Q + ((size_t)bh * SEQ + nrow) * HD + 2 * hi;

    v8f acc[4];
#pragma unroll
    for (int t = 0; t < 4; ++t) acc[t] = (v8f){};

    for (int k = 0; k < HD; k += 4) {
        v2f a = *(const v2f*)(qp + k);
#pragma unroll
        for (int t = 0; t < 4; ++t) {
            v2f b;
            b.x = sMap[(k + 2 * hi) * 64 + t * 16 + lx];
            b.y = sMap[(k + 2 * hi + 1) * 64 + t * 16 + lx];
            acc[t] = wmma_f32(a, b, acc[t]);
        }
    }

    const int b = bh >> 4, h = bh & 15;
#pragma unroll
    for (int t = 0; t < 4; ++t) {
        const int ecol = t * 16 + lx;
#pragma unroll
        for (int r = 0; r < 8; ++r) {
            const int n = blockIdx.y * 128 + wave * 16 + r + 8 * hi;
            outbnc[(((size_t)b * SEQ) + n) * CDIM + h * HD + ecol] = acc[t][r];
        }
    }
}

// ---------------------------------------------------------------------------
// Host launcher
// ---------------------------------------------------------------------------
extern "C" void kernel_launch(void* const* d_in, const int* in_sizes, int n_in,
                              void* d_out, int out_size, void* d_ws, size_t ws_size,
                              hipStream_t stream)
{
    (void)in_sizes; (void)n_in; (void)out_size; (void)ws_size;
    const float* x     = (const float*)d_in[0];
    const float* wq    = (const float*)d_in[1];
    const float* wk    = (const float*)d_in[2];
    const float* wv    = (const float*)d_in[3];
    const float* ow    = (const float*)d_in[4];
    const float* ob    = (const float*)d_in[5];
    const float* alpha = (const float*)d_in[6];

    float* out  = (float*)d_out;                       // [8,1024,1024]
    float* trad = out + (size_t)BDIM * SEQ * CDIM;     // [8,16,1024,1024]

    // Workspace layout (floats): Q, K, V [BH][SEQ][HD]; maps [BH][64][64];
    // attn [B][N][C].  Total ~130 MB.
    float* Qw   = (float*)d_ws;
    float* Kw   = Qw + (size_t)BH * SEQ * HD;
    float* Vw   = Kw + (size_t)BH * SEQ * HD;
    float* maps = Vw + (size_t)BH * SEQ * HD;
    float* attn = maps + (size_t)BH * HD * HD;

    const int M = BDIM * SEQ;  // 8192

    // 1) Q/K/V projections (headed-layout scatter)
    dim3 ggrid(M / 256, CDIM / 64);
    gemm_nt_kernel<<<ggrid, 256, 0, stream>>>(x, wq, nullptr, Qw, M, CDIM, CDIM, 1);
    gemm_nt_kernel<<<ggrid, 256, 0, stream>>>(x, wk, nullptr, Kw, M, CDIM, CDIM, 1);
    gemm_nt_kernel<<<ggrid, 256, 0, stream>>>(x, wv, nullptr, Vw, M, CDIM, CDIM, 1);

    // 2) trad_map = softmax_q(Q K^T / 8)
    scores_kernel<<<dim3(BH, SEQ / 256, SEQ / 64), 256, 0, stream>>>(Qw, Kw, trad);
    colsoftmax_kernel<<<dim3(BH, SEQ / 256), 256, 0, stream>>>(trad);

    // 3) per-(b,h) regression map: softmax_d((alpha*I + K^T K)^-1 K^T V)
    solve_kernel<<<dim3(BH), 256, 0, stream>>>(Kw, Vw, alpha, maps);

    // 4) attn = Q @ map, scattered to [b, n, c]
    qmap_kernel<<<dim3(BH, SEQ / 128), 256, 0, stream>>>(Qw, maps, attn);

    // 5) out = attn @ out_w^T + out_b
    gemm_nt_kernel<<<ggrid, 256, 0, stream>>>(attn, ow, ob, out, M, CDIM, CDIM, 0);
}